// MultiSamplePatches_57148834841008
// MI455X (gfx1250) — compile-verified
//
#include <hip/hip_runtime.h>
#include <stdint.h>

// ---- problem constants (from reference) ----
#define B_      8
#define HL_     112
#define WL_     112
#define HW_     (HL_*WL_)          // 12544 attention entries per batch
#define HH_     1024
#define WH_     1024
#define PATCH_  100
#define NP_     10
#define ROWF_   (PATCH_*3)         // 300 floats per patch row
#define PATCH_F (PATCH_*PATCH_*3)  // 30000 floats per patch
#define SRC_ROW (WH_*3)            // 3072 floats per source row
#define HI_MAX  (HH_ - PATCH_)     // 924
#define TOTAL_U (B_*HW_)           // 100352 uniforms
#define HALF_U  (TOTAL_U/2)        // 50176

typedef unsigned int v4u __attribute__((ext_vector_type(4)));
typedef int          v8i __attribute__((ext_vector_type(8)));
typedef int          v4i __attribute__((ext_vector_type(4)));

// ---------------- JAX threefry2x32-20, key = jax.random.key(1) => (k0,k1)=(0,1) -------------
__device__ __forceinline__ uint32_t rotl32(uint32_t x, int r) {
  return (x << r) | (x >> (32 - r));
}

__device__ float jax_uniform01(uint32_t gi) {
  const uint32_t k0 = 0u, k1 = 1u;
  const uint32_t k2 = 0x1BD11BDAu ^ k0 ^ k1;
  uint32_t x0, x1;
  const bool lo = gi < HALF_U;               // counts split: out[i]=word0, out[i+half]=word1
  if (lo) { x0 = gi;           x1 = gi + HALF_U; }
  else    { x0 = gi - HALF_U;  x1 = gi;          }
  x0 += k0; x1 += k1;
#define TFR(r) { x0 += x1; x1 = rotl32(x1, (r)); x1 ^= x0; }
  TFR(13) TFR(15) TFR(26) TFR(6)   x0 += k1; x1 += k2 + 1u;
  TFR(17) TFR(29) TFR(16) TFR(24)  x0 += k2; x1 += k0 + 2u;
  TFR(13) TFR(15) TFR(26) TFR(6)   x0 += k0; x1 += k1 + 3u;
  TFR(17) TFR(29) TFR(16) TFR(24)  x0 += k1; x1 += k2 + 4u;
  TFR(13) TFR(15) TFR(26) TFR(6)   x0 += k2; x1 += k0 + 5u;
#undef TFR
  const uint32_t bits = lo ? x0 : x1;
  return __uint_as_float((bits >> 9) | 0x3F800000u) - 1.0f;   // [0,1), matches jax.random.uniform f32
}

// ---------------- Kernel 1: Gumbel top-k sampling + coordinate routing ----------------
__global__ __launch_bounds__(256)
void sample_topk_kernel(const float* __restrict__ ats_map,
                        const int*   __restrict__ ats_index,
                        const float* __restrict__ scales,
                        float* __restrict__ out_att,
                        float* __restrict__ out_offsets,
                        float* __restrict__ out_samples,
                        long long* __restrict__ ws_src)
{
  __shared__ float sp[HW_];        // raw attention, then perturbed logits (in place)
  __shared__ float red[256];
  __shared__ int   redi[256];
  __shared__ int   sel[NP_];
  __shared__ float s_inv;

  const int b   = blockIdx.x;
  const int tid = threadIdx.x;
  const float* amrow = ats_map + (long long)b * HW_;

  // row sum (for normalization)
  float acc = 0.f;
  for (int i = tid; i < HW_; i += 256) { float a = amrow[i]; sp[i] = a; acc += a; }
  red[tid] = acc; __syncthreads();
  for (int s = 128; s > 0; s >>= 1) { if (tid < s) red[tid] += red[tid + s]; __syncthreads(); }
  if (tid == 0) s_inv = 1.0f / red[0];
  __syncthreads();
  const float inv = s_inv;

  // perturbed logits: log(att/sum + 1e-10) + gumbel(u)
  for (int i = tid; i < HW_; i += 256) {
    const float u = jax_uniform01((uint32_t)(b * HW_ + i));
    const float g = -logf(-logf(u + 1e-10f) + 1e-10f);
    sp[i] = logf(sp[i] * inv + 1e-10f) + g;
  }
  __syncthreads();

  // iterative top-10 (descending, ties -> lowest index, like lax.top_k)
  for (int k = 0; k < NP_; ++k) {
    float bv = -INFINITY; int bi = 0x7fffffff;
    for (int i = tid; i < HW_; i += 256) {
      const float v = sp[i];
      if (v > bv) { bv = v; bi = i; }
    }
    red[tid] = bv; redi[tid] = bi; __syncthreads();
    for (int s = 128; s > 0; s >>= 1) {
      if (tid < s) {
        const float ov = red[tid + s]; const int oi = redi[tid + s];
        if (ov > red[tid] || (ov == red[tid] && oi < redi[tid])) { red[tid] = ov; redi[tid] = oi; }
      }
      __syncthreads();
    }
    if (tid == 0) { sel[k] = redi[0]; sp[redi[0]] = -INFINITY; }
    __syncthreads();
  }

  // per-sample outputs + gather descriptors
  if (tid < NP_) {
    const int n   = tid;
    const int idx = sel[n];
    const float sy = (float)(idx / WL_);
    const float sx = (float)(idx % WL_);
    const int   s  = ats_index[(long long)b * HW_ + idx];
    const float scl = scales[s];
    const float cy = sy / scl, cx = sx / scl;
    const float ratio = (float)(HH_ - PATCH_) / (float)(HL_ - 1);   // 924/111 (both dims)
    int ry = (int)rintf(cy * ratio); ry = ry < 0 ? 0 : (ry > HI_MAX ? HI_MAX : ry);
    int rx = (int)rintf(cx * ratio); rx = rx < 0 ? 0 : (rx > HI_MAX ? HI_MAX : rx);
    const int o = b * NP_ + n;
    out_att[o]            = amrow[idx] * inv;
    out_offsets[2*o]      = 0.0f;      // RECEPTIVE_FIELD/2 == 0
    out_offsets[2*o + 1]  = 0.0f;
    out_samples[2*o]      = cy;
    out_samples[2*o + 1]  = cx;
    ws_src[o] = (long long)s * ((long long)B_ * HH_ * WH_ * 3)
              + (long long)b * ((long long)HH_ * WH_ * 3)
              + (long long)ry * SRC_ROW + (long long)rx * 3;
  }
}

// ---------------- Kernel 2: patch gather = two TDM descriptors (pure DMA) ----------------
__device__ __forceinline__ uint32_t lds_byte_off(const void* p) {
  // generic -> LDS addrspace -> raw LDS byte address
  return (uint32_t)(uintptr_t)(__attribute__((address_space(3))) const void*)p;
}

// Build D# group 0: count=1 | lds_addr | global_addr[56:0] | type=2 ("image")
__device__ __forceinline__ v4u tdm_group0(uint32_t lds_addr, uint64_t gaddr) {
  v4u g;
  g[0] = 1u;                                             // count=1, user mode, no gather
  g[1] = lds_addr;                                       // lds_addr (bytes)
  g[2] = (unsigned int)(gaddr & 0xFFFFFFFFull);          // global_addr[31:0]
  g[3] = (unsigned int)((gaddr >> 32) & 0x01FFFFFFull)   // global_addr[56:32]
       | 0x80000000u;                                    // type=2
  return g;
}

// Build D# group 1: data_size=4B, 2D tile (tile_dim0 x tile_dim1), row stride in elements.
// Tensor dims set huge: corners are pre-clipped so the tile never leaves the image.
__device__ __forceinline__ v8i tdm_group1(int tile_d0, int tile_d1, int stride_d0) {
  v8i g;
  g[0] = 0x00020000;                          // workgroup_mask=0 | data_size=2(4B), no pad/iterate
  g[1] = (int)0xFFFF0000;                     // barrier_addr=0 | tensor_dim0[15:0]
  g[2] = (int)0xFFFF7FFF;                     // tensor_dim0[31:16]=0x7FFF | tensor_dim1[15:0]
  g[3] = (int)((tile_d0 << 16) | 0x7FFF);     // tensor_dim1[31:16]=0x7FFF | tile_dim0
  g[4] = tile_d1;                             // tile_dim1 | tile_dim2=0
  g[5] = stride_d0;                           // tensor_dim0_stride[31:0] (elements)
  g[6] = 0;                                   // stride[47:32]=0 | tensor_dim1_stride[15:0]=0
  g[7] = 0;                                   // tensor_dim1_stride[47:16]=0
  return g;
}

__global__ __launch_bounds__(32)            // one wave per patch: the TDM does all data movement
void gather_patches_tdm_kernel(const float* __restrict__ x_highs,
                               const long long* __restrict__ ws_src,
                               float* __restrict__ out_patches)
{
  extern __shared__ __align__(16) float tile[];   // 30000 floats = 120000 B dynamic LDS staging
  const int bn = blockIdx.x;                      // b*NP + n  (uniform)
  const long long srcElem = ws_src[bn];           // uniform -> scalar load
  const uint64_t src_addr = (uint64_t)(const void*)x_highs + (uint64_t)srcElem * 4ull;
  const uint64_t dst_addr = (uint64_t)(const void*)(out_patches + (long long)bn * PATCH_F);
  const uint32_t ldsb     = lds_byte_off(tile);

  const v4i gz4 = {0, 0, 0, 0};                   // groups 2/3 unused (2D tensors)
  const v8i gz8 = {0, 0, 0, 0, 0, 0, 0, 0};

  // ---- TDM load: 100 rows x 300 f32 strided tile (stride 3072 f32) -> LDS ----
  {
    const v4u g0 = tdm_group0(ldsb, src_addr);
    const v8i g1 = tdm_group1(ROWF_, PATCH_, SRC_ROW);
    __builtin_amdgcn_tensor_load_to_lds(g0, g1, gz4, gz4, gz8, 0);
  }
  // Fence between the two descriptors: inter-descriptor ordering is only guaranteed via the
  // atomic-barrier option (ISA 9.1), so drain TENSORcnt before the store reads the LDS tile.
  __builtin_amdgcn_s_wait_tensorcnt(0);

  // ---- TDM store: LDS -> contiguous output patch (300 x 100 tile, row stride 300) ----
  {
    const v4u g0 = tdm_group0(ldsb, dst_addr);
    const v8i g1 = tdm_group1(ROWF_, PATCH_, ROWF_);
    __builtin_amdgcn_tensor_store_from_lds(g0, g1, gz4, gz4, gz8, 0);
  }
  __builtin_amdgcn_s_wait_tensorcnt(0);           // drain before LDS dealloc / wave end
}

// ---------------- host wrapper ----------------
extern "C" void kernel_launch(void* const* d_in, const int* in_sizes, int n_in,
                              void* d_out, int out_size, void* d_ws, size_t ws_size,
                              hipStream_t stream) {
  // inputs (setup_inputs order): x_lows [unused], x_highs, ats_map, ats_index, scales
  const float* x_highs   = (const float*)d_in[1];
  const float* ats_map   = (const float*)d_in[2];
  const int*   ats_index = (const int*)  d_in[3];
  const float* scales    = (const float*)d_in[4];

  float* out          = (float*)d_out;
  float* out_patches  = out;                         // 8*10*100*100*3 = 2,400,000
  float* out_att      = out + 2400000;               // 80
  float* out_offsets  = out + 2400080;               // 160
  float* out_samples  = out + 2400240;               // 160
  long long* ws_src   = (long long*)d_ws;            // 80 source element offsets

  hipLaunchKernelGGL(sample_topk_kernel, dim3(B_), dim3(256), 0, stream,
                     ats_map, ats_index, scales,
                     out_att, out_offsets, out_samples, ws_src);
  // 120000 B dynamic LDS staging tile per workgroup (320 KB/WGP available on CDNA5)
  hipLaunchKernelGGL(gather_patches_tdm_kernel, dim3(B_ * NP_), dim3(32),
                     PATCH_F * sizeof(float), stream,
                     x_highs, ws_src, out_patches);
}